// cross_attention_kitti_64604898067014
// MI455X (gfx1250) — compile-verified
//
#include <hip/hip_runtime.h>
#include <hip/hip_bf16.h>
#include <stdint.h>

// ---------------- problem constants (from reference) ----------------
#define BS     2
#define NH_    64
#define NV_    8
#define HF_    26
#define WF_    88
#define C_     256
#define HEADS_ 8
#define P_     4
#define DH_    32              // C/HEADS
#define NU_    33              // NH/2+1
#define LQ_    2112            // NU*NH
#define IMGW_  1408.0f
#define IMGH_  416.0f
#define GH_    51.2f
#define GV_    8.0f

// output layout in d_out (floats): grd_bev | max_height_index | keep_index
#define OUT_GRD_OFF  0
#define OUT_IDX_OFF  (BS * LQ_ * C_)                 // 1,081,344
#define OUT_KEEP_OFF (OUT_IDX_OFF + BS * NU_ * NH_)  // +4,224

typedef __attribute__((ext_vector_type(2))) float v2f;
typedef __attribute__((ext_vector_type(8))) float v8f;

// ---------------------------------------------------------------------------
// Kernel 1: projective grid -> ref (u,v) per (b, z, q) and keep_index
// ---------------------------------------------------------------------------
__global__ void k_proj(const float* __restrict__ camk,
                       float* __restrict__ ref_uv,     // (BS, NV, LQ, 2)
                       float* __restrict__ keep_out)   // (BS, NU*NH) as 0/1
{
    int t = blockIdx.x * blockDim.x + threadIdx.x;
    if (t >= BS * NU_ * NH_) return;
    int ih = t % NH_;
    int iu = (t / NH_) % NU_;
    int b  = t / (NH_ * NU_);
    const float* K = camk + b * 9;

    float x = -GH_ * 0.5f + iu * ((GH_ * 0.5f) / (NU_ - 1)); // linspace(-GH/2, 0, NU)
    float y = -GH_ * 0.5f + ih * (GH_ / (NH_ - 1));          // linspace(-GH/2, GH/2, NH)
    int q = iu * NH_ + ih;

    bool keep = false;
    for (int z = 0; z < NV_; ++z) {
        float zz = -GV_ * 0.5f + z * (GV_ / (NV_ - 1));
        // g = (y, -z, -x)
        float g0 = y, g1 = -zz, g2 = -x;
        float p0 = K[0] * g0 + K[1] * g1 + K[2] * g2;
        float p1 = K[3] * g0 + K[4] * g1 + K[5] * g2;
        float p2 = K[6] * g0 + K[7] * g1 + K[8] * g2;
        float den = (fabsf(p2) < 1e-6f) ? 1e-6f : p2;
        float u = p0 / den / IMGW_;
        float v = p1 / den / IMGH_;
        keep = keep || (u >= 0.f && u <= 1.f && v >= 0.f && v <= 1.f);
        size_t o = ((size_t)(b * NV_ + z) * LQ_ + q) * 2;
        ref_uv[o + 0] = u;
        ref_uv[o + 1] = v;
    }
    keep_out[b * NU_ * NH_ + q] = keep ? 1.0f : 0.0f;
}

// ---------------------------------------------------------------------------
// Kernel 2: f32 WMMA GEMM, K fixed at 256: Out[M,N] = A[M,256] @ W[256,N] + b
//
// Block = 256 threads = 8 waves. Each block owns ONE 16-wide N panel (tn) and
// eight consecutive M tiles. The 256x16 f32 weight panel (16 KB) is staged
// into LDS once per block with GLOBAL_LOAD_ASYNC_TO_LDS_B128 (ASYNCcnt path,
// no VGPR round-trip), then all 8 waves stream it from LDS -> 8x less global
// B traffic and 1 global load (A b64) per V_WMMA_F32_16X16X4_F32.
//
// Fragment layouts per ISA 7.12.2:
//   A (16x4): lane&15 = row, lane>>4 selects K pair {0,1}/{2,3} in v2f
//   B (4x16): lane&15 = col, lane>>4 selects K pair
//   C/D     : vgpr i -> row (lane>>4)*8+i, lane&15 -> col
// ---------------------------------------------------------------------------
#define GEMM_K 256

__global__ void k_gemm_wmma_f32(const float* __restrict__ A,
                                const float* __restrict__ W,
                                const float* __restrict__ bias,
                                float* __restrict__ Out,
                                int M, int N,
                                const float* __restrict__ res,  // nullptr -> none
                                int res_chunk)                  // NV*LQ for grd gemm
{
    __shared__ float bpan[GEMM_K * 16];   // W[k][tn*16 + 0..15], 16 KB

    int tiles_n = N >> 4;
    int tiles_m = M >> 4;
    int tn      = blockIdx.x % tiles_n;
    int tm      = (blockIdx.x / tiles_n) * 8 + (threadIdx.x >> 5);

    // ---- stage B panel: 4096 floats, 256 threads x 4 iters x 16B async ----
    {
        int tid = threadIdx.x;
        for (int e = tid * 4; e < GEMM_K * 16; e += 256 * 4) {
            int k    = e >> 4;
            int cofs = e & 15;
            const float* src = W + (size_t)k * N + tn * 16 + cofs;
            unsigned lds = (unsigned)(uintptr_t)(&bpan[e]);
            uint64_t ga  = (uint64_t)(uintptr_t)src;
            asm volatile("global_load_async_to_lds_b128 %0, %1, off"
                         :: "v"(lds), "v"(ga) : "memory");
        }
        asm volatile("s_wait_asynccnt 0x0" ::: "memory");
    }
    __syncthreads();

    if (tm < tiles_m) {
        int lane = threadIdx.x & 31;
        int half = lane >> 4;   // 0: K{0,1}, 1: K{2,3}
        int l    = lane & 15;

        const float* arow = A + (size_t)(tm * 16 + l) * GEMM_K + half * 2;
        const float* bl   = &bpan[half * 2 * 16 + l];

        v8f acc = {};
        for (int kk = 0; kk < GEMM_K; kk += 4) {
            v2f a;  a.x = arow[0];  a.y = arow[1];     // one b64 global load
            v2f bf; bf.x = bl[0];   bf.y = bl[16];     // two ds_load_b32
            acc = __builtin_amdgcn_wmma_f32_16x16x4_f32(
                /*neg_a=*/false, a, /*neg_b=*/false, bf,
                /*c_mod=*/(short)0, acc, /*reuse_a=*/false, /*reuse_b=*/false);
            arow += 4;
            bl   += 64;
        }

        int   col = tn * 16 + l;
        float bvv = bias ? bias[col] : 0.0f;
#pragma unroll
        for (int i = 0; i < 8; ++i) {
            int row = tm * 16 + half * 8 + i;
            float v = acc[i] + bvv;
            if (res) {
                int bb = row / res_chunk;   // batch
                int q  = row % LQ_;         // query index
                v += res[((size_t)bb * LQ_ + q) * N + col];
            }
            Out[(size_t)row * N + col] = v;
        }
    }
}

// ---------------------------------------------------------------------------
// Kernel 3: softmax over P=4 for attention weights
// logits layout: (BS*LQ, HEADS*P) row-major -> flat t*P for t=(b*LQ+q)*HEADS+h
// ---------------------------------------------------------------------------
__global__ void k_attn_softmax(const float* __restrict__ logits,
                               float* __restrict__ attn)
{
    int t = blockIdx.x * blockDim.x + threadIdx.x;
    if (t >= BS * LQ_ * HEADS_) return;
    const float* L = logits + (size_t)t * P_;
    float m = fmaxf(fmaxf(L[0], L[1]), fmaxf(L[2], L[3]));
    float e0 = __expf(L[0] - m), e1 = __expf(L[1] - m);
    float e2 = __expf(L[2] - m), e3 = __expf(L[3] - m);
    float inv = 1.0f / (e0 + e1 + e2 + e3);
    float* O = attn + (size_t)t * P_;
    O[0] = e0 * inv; O[1] = e1 * inv; O[2] = e2 * inv; O[3] = e3 * inv;
}

// ---------------------------------------------------------------------------
// Kernel 4: bilinear sampling + attention-weighted sum -> msda (BS,NV,LQ,C)
// One wave per (b, v, q, h); 32 lanes = DH channels (coalesced 128B per corner)
// ---------------------------------------------------------------------------
__device__ __forceinline__ float bilin_corner(const float* __restrict__ vb,
                                              int xi, int yi, float w)
{
    bool valid = (xi >= 0) && (xi < WF_) && (yi >= 0) && (yi < HF_);
    int xc = min(max(xi, 0), WF_ - 1);
    int yc = min(max(yi, 0), HF_ - 1);
    float g = vb[((size_t)yc * WF_ + xc) * C_];
    return valid ? g * w : 0.0f;
}

__global__ void k_sample(const float* __restrict__ valp,   // (BS,HF,WF,HEADS,DH)
                         const float* __restrict__ ref_uv, // (BS,NV,LQ,2)
                         const float* __restrict__ off,    // (BS*LQ, HEADS*P*2)
                         const float* __restrict__ attn,   // (BS*LQ, HEADS*P)
                         float* __restrict__ msda)         // (BS,NV,LQ,C)
{
    int wave = blockIdx.x * (blockDim.x >> 5) + (threadIdx.x >> 5);
    if (wave >= BS * NV_ * LQ_ * HEADS_) return;
    int h = wave % HEADS_;
    int q = (wave / HEADS_) % LQ_;
    int v = (wave / (HEADS_ * LQ_)) % NV_;
    int b = wave / (HEADS_ * LQ_ * NV_);
    int d = threadIdx.x & 31;

    size_t ro = ((size_t)(b * NV_ + v) * LQ_ + q) * 2;
    float u0 = ref_uv[ro + 0];
    float v0 = ref_uv[ro + 1];

    const float* offq = off  + (size_t)(b * LQ_ + q) * (HEADS_ * P_ * 2) + h * P_ * 2;
    const float* attq = attn + (size_t)(b * LQ_ + q) * (HEADS_ * P_)     + h * P_;
    const float* vb   = valp + (size_t)b * HF_ * WF_ * C_ + h * DH_ + d;

    float acc = 0.0f;
#pragma unroll
    for (int p = 0; p < P_; ++p) {
        float px = u0 * WF_ + offq[p * 2 + 0] - 0.5f;
        float py = v0 * HF_ + offq[p * 2 + 1] - 0.5f;
        px = fminf(fmaxf(px, -10000.0f), 10000.0f);
        py = fminf(fmaxf(py, -10000.0f), 10000.0f);
        float x0f = floorf(px), y0f = floorf(py);
        float lx = px - x0f, ly = py - y0f;
        int x0 = (int)x0f, y0 = (int)y0f;
        float s = bilin_corner(vb, x0,     y0,     (1.f - lx) * (1.f - ly))
                + bilin_corner(vb, x0 + 1, y0,     lx * (1.f - ly))
                + bilin_corner(vb, x0,     y0 + 1, (1.f - lx) * ly)
                + bilin_corner(vb, x0 + 1, y0 + 1, lx * ly);
        acc += attq[p] * s;
    }
    msda[((size_t)(b * NV_ + v) * LQ_ + q) * C_ + h * DH_ + d] = acc;
}

// ---------------------------------------------------------------------------
// Kernel 5: height softmax over NV + argmax + weighted sum + residual
// One block (256 threads = 8 waves) per (b, q). Wave z computes dot with Wp.
// ---------------------------------------------------------------------------
__global__ void k_height(const float* __restrict__ grd,    // (BS,NV,LQ,C)
                         const float* __restrict__ query,  // (BS,LQ,C)
                         const float* __restrict__ Wp,     // (C,1)
                         const float* __restrict__ bp,     // (1,)
                         float* __restrict__ out_grd,      // (BS,LQ,C)
                         float* __restrict__ out_idx)      // (BS,NU*NH)
{
    int bq = blockIdx.x;               // 0 .. BS*LQ-1
    int b = bq / LQ_, q = bq % LQ_;
    int tid = threadIdx.x;             // 0..255
    int z = tid >> 5, lane = tid & 31;

    __shared__ float s_dot[NV_];
    __shared__ float s_w[NV_];
    __shared__ int   s_arg;

    const float* row = grd + ((size_t)(b * NV_ + z) * LQ_ + q) * C_;
    float part = 0.0f;
    for (int c = lane; c < C_; c += 32) part += row[c] * Wp[c];
#pragma unroll
    for (int o = 16; o > 0; o >>= 1) part += __shfl_xor(part, o, 32);
    if (lane == 0) s_dot[z] = part + bp[0];
    __syncthreads();

    if (tid == 0) {
        float m = s_dot[0]; int arg = 0;
        for (int i = 1; i < NV_; ++i)
            if (s_dot[i] > m) { m = s_dot[i]; arg = i; }
        float e[NV_]; float sum = 0.0f;
        for (int i = 0; i < NV_; ++i) { e[i] = __expf(s_dot[i] - m); sum += e[i]; }
        float inv = 1.0f / sum;
        for (int i = 0; i < NV_; ++i) s_w[i] = e[i] * inv;
        s_arg = arg;
    }
    __syncthreads();

    int c = tid;  // blockDim == C
    float accum = 0.0f;
#pragma unroll
    for (int zz = 0; zz < NV_; ++zz)
        accum += s_w[zz] * grd[((size_t)(b * NV_ + zz) * LQ_ + q) * C_ + c];
    out_grd[(size_t)bq * C_ + c] = accum + query[(size_t)bq * C_ + c];
    if (tid == 0) out_idx[b * NU_ * NH_ + q] = (float)s_arg;
}

// ---------------------------------------------------------------------------
// Launch
// ---------------------------------------------------------------------------
static inline int cdiv(int a, int b) { return (a + b - 1) / b; }

extern "C" void kernel_launch(void* const* d_in, const int* in_sizes, int n_in,
                              void* d_out, int out_size, void* d_ws, size_t ws_size,
                              hipStream_t stream)
{
    const float* query = (const float*)d_in[0];   // (2, 2112, 256)
    const float* value = (const float*)d_in[1];   // (2, 2288, 256)
    const float* camk  = (const float*)d_in[2];   // (2, 3, 3)
    const float* Wv    = (const float*)d_in[3];
    const float* bv    = (const float*)d_in[4];
    const float* Wo    = (const float*)d_in[5];
    const float* bo    = (const float*)d_in[6];
    const float* Wa    = (const float*)d_in[7];
    const float* ba    = (const float*)d_in[8];
    const float* Wout  = (const float*)d_in[9];
    const float* bout  = (const float*)d_in[10];
    const float* Wp    = (const float*)d_in[11];
    const float* bp    = (const float*)d_in[12];

    float* out = (float*)d_out;

    // workspace partition (floats)
    float* ws       = (float*)d_ws;
    float* ws_ref   = ws;                                      // BS*NV*LQ*2     =    67,584
    float* ws_valp  = ws_ref   + (size_t)BS * NV_ * LQ_ * 2;   // BS*HF*WF*C     = 1,171,456
    float* ws_off   = ws_valp  + (size_t)BS * HF_ * WF_ * C_;  // BS*LQ*64       =   270,336
    float* ws_logit = ws_off   + (size_t)BS * LQ_ * 64;        // BS*LQ*32       =   135,168
    float* ws_attn  = ws_logit + (size_t)BS * LQ_ * 32;        // BS*LQ*32       =   135,168
    float* ws_msda  = ws_attn  + (size_t)BS * LQ_ * 32;        // BS*NV*LQ*C     = 8,650,752
    float* ws_grd   = ws_msda  + (size_t)BS * NV_ * LQ_ * C_;  // BS*NV*LQ*C     = 8,650,752

    // 1. projection / ref grid / keep_index
    {
        int n = BS * NU_ * NH_;
        k_proj<<<cdiv(n, 256), 256, 0, stream>>>(camk, ws_ref, out + OUT_KEEP_OFF);
    }
    // 2. valp = value @ Wv + bv   (4576 x 256 x 256)
    {
        int M = BS * HF_ * WF_, N = C_;
        int blocks = (N / 16) * cdiv(M / 16, 8);
        k_gemm_wmma_f32<<<blocks, 256, 0, stream>>>(
            value, Wv, bv, ws_valp, M, N, nullptr, 1);
    }
    // 3. off = query @ Wo + bo   (4224 x 256 x 64)
    {
        int M = BS * LQ_, N = HEADS_ * P_ * 2;
        int blocks = (N / 16) * cdiv(M / 16, 8);
        k_gemm_wmma_f32<<<blocks, 256, 0, stream>>>(
            query, Wo, bo, ws_off, M, N, nullptr, 1);
    }
    // 4. attn logits = query @ Wa + ba   (4224 x 256 x 32)
    {
        int M = BS * LQ_, N = HEADS_ * P_;
        int blocks = (N / 16) * cdiv(M / 16, 8);
        k_gemm_wmma_f32<<<blocks, 256, 0, stream>>>(
            query, Wa, ba, ws_logit, M, N, nullptr, 1);
    }
    // 5. softmax over P
    {
        int n = BS * LQ_ * HEADS_;
        k_attn_softmax<<<cdiv(n, 256), 256, 0, stream>>>(ws_logit, ws_attn);
    }
    // 6. bilinear sampling + attn-weighted sum -> msda
    {
        int waves = BS * NV_ * LQ_ * HEADS_;   // 270,336
        k_sample<<<cdiv(waves, 8), 256, 0, stream>>>(
            ws_valp, ws_ref, ws_off, ws_attn, ws_msda);
    }
    // 7. grd = msda @ Wout + bout + query[:,None]   (33792 x 256 x 256)
    {
        int M = BS * NV_ * LQ_, N = C_;
        int blocks = (N / 16) * cdiv(M / 16, 8);
        k_gemm_wmma_f32<<<blocks, 256, 0, stream>>>(
            ws_msda, Wout, bout, ws_grd, M, N, query, NV_ * LQ_);
    }
    // 8. height softmax + argmax + weighted sum + residual
    {
        k_height<<<BS * LQ_, 256, 0, stream>>>(
            ws_grd, query, Wp, bp, out + OUT_GRD_OFF, out + OUT_IDX_OFF);
    }
}